// GCN_7060926234753
// MI455X (gfx1250) — compile-verified
//
#include <hip/hip_runtime.h>

typedef float v2f __attribute__((ext_vector_type(2)));
typedef float v8f __attribute__((ext_vector_type(8)));

#define Nn 50000
#define Ee 600000
#define Ff 128
#define Hh 128
#define Cc 10
#define Gg 64

// ---------------- degree / normalization ----------------

__global__ void deg_count_kernel(const int* __restrict__ dst, int* __restrict__ degc, int E) {
    int i = blockIdx.x * blockDim.x + threadIdx.x;
    if (i < E) atomicAdd(&degc[dst[i]], 1);
}

__global__ void dinv_kernel(const int* __restrict__ degc, float* __restrict__ dinv, int N) {
    int i = blockIdx.x * blockDim.x + threadIdx.x;
    if (i < N) {
        float d = (float)(degc[i] + 1);   // +1 self-loop
        dinv[i] = rsqrtf(d);              // deg always >= 1
    }
}

__global__ void norm_kernel(const int* __restrict__ src, const int* __restrict__ dst,
                            const float* __restrict__ dinv, float* __restrict__ norm, int E) {
    int i = blockIdx.x * blockDim.x + threadIdx.x;
    if (i < E) norm[i] = dinv[src[i]] * dinv[dst[i]];
}

// ---------------- fp32 WMMA GEMM: Y[N,128] = X[N,128] @ W[128,128] ----------------
// One 256-thread block = 8 waves; wave w owns output columns [16w, 16w+16).
// Block handles 5 row-tiles of 16 rows (625 blocks * 80 rows = 50000 exactly).

__global__ __launch_bounds__(256) void gemm_wmma_f32(const float* __restrict__ X,
                                                     const float* __restrict__ W,
                                                     float* __restrict__ Y) {
    __shared__ float sW[Hh * Ff];  // 64 KB, row-major [k][n]
    {
        const float4* Wv = (const float4*)W;
        float4* sWv = (float4*)sW;
        #pragma unroll
        for (int i = threadIdx.x; i < (Hh * Ff) / 4; i += 256) sWv[i] = Wv[i];
    }
    __syncthreads();

    const int lane = threadIdx.x & 31;
    const int wave = threadIdx.x >> 5;
    const int m15  = lane & 15;
    const int half = lane >> 4;          // 0: K pair {0,1}, 1: K pair {2,3}
    const int colg = wave * 16 + m15;    // this lane's output column

    for (int mt = 0; mt < 5; ++mt) {
        const int row0 = (blockIdx.x * 5 + mt) * 16;
        const float* xrow = X + (size_t)(row0 + m15) * Ff + half * 2;
        v8f acc = {0.f, 0.f, 0.f, 0.f, 0.f, 0.f, 0.f, 0.f};
        #pragma unroll
        for (int k0 = 0; k0 < Ff; k0 += 4) {
            v2f a;                        // A 16x4: lane(m15) row, K = k0 + half*2 + {0,1}
            a.x = xrow[k0];
            a.y = xrow[k0 + 1];
            const int kp = k0 + half * 2; // B 4x16: K rows striped like C/D
            v2f b;
            b.x = sW[kp * Hh + colg];
            b.y = sW[(kp + 1) * Hh + colg];
            acc = __builtin_amdgcn_wmma_f32_16x16x4_f32(
                false, a, false, b, (short)0, acc, false, false);
        }
        // C/D layout: VGPR r -> M = r (lanes 0-15), M = r+8 (lanes 16-31), N = lane&15
        #pragma unroll
        for (int r = 0; r < 8; ++r) {
            Y[(size_t)(row0 + half * 8 + r) * Hh + colg] = acc[r];
        }
    }
}

// ---------------- edge scatter: agg[dst] += norm[e] * Y[src] ----------------
// One wave per edge; lane handles features [4*lane, 4*lane+4).

__global__ __launch_bounds__(256) void edge_scatter_kernel(const float* __restrict__ Y,
                                                           const int* __restrict__ src,
                                                           const int* __restrict__ dst,
                                                           const float* __restrict__ norm,
                                                           float* __restrict__ agg, int E) {
    int gid  = blockIdx.x * blockDim.x + threadIdx.x;
    int e    = gid >> 5;
    int lane = gid & 31;
    if (e >= E) return;
    int   s  = src[e];
    int   d  = dst[e];
    float nv = norm[e];
    float4 v = ((const float4*)(Y + (size_t)s * Hh))[lane];
    float* ap = agg + (size_t)d * Hh + lane * 4;
    unsafeAtomicAdd(ap + 0, v.x * nv);
    unsafeAtomicAdd(ap + 1, v.y * nv);
    unsafeAtomicAdd(ap + 2, v.z * nv);
    unsafeAtomicAdd(ap + 3, v.w * nv);
}

// ---------------- self-loop + bias + optional ReLU (in-place on agg) ----------------

__global__ void fuse_kernel(float* __restrict__ agg, const float* __restrict__ Y,
                            const float* __restrict__ dinv, const float* __restrict__ bias,
                            int doRelu) {
    int gid  = blockIdx.x * blockDim.x + threadIdx.x;  // N*H threads
    int node = gid >> 7;
    int f    = gid & 127;
    float di = dinv[node];
    float v  = agg[gid] + di * di * Y[gid] + bias[f];
    if (doRelu) v = fmaxf(v, 0.f);
    agg[gid] = v;
}

// ---------------- global mean pool ----------------

__global__ __launch_bounds__(256) void pool_kernel(const float* __restrict__ h,
                                                   const int* __restrict__ batch,
                                                   float* __restrict__ pooled) {
    int gid  = blockIdx.x * blockDim.x + threadIdx.x;
    int node = gid >> 5;
    int lane = gid & 31;
    if (node >= Nn) return;
    int g = batch[node];
    float4 v = ((const float4*)(h + (size_t)node * Hh))[lane];
    float* pp = pooled + (size_t)g * Hh + lane * 4;
    unsafeAtomicAdd(pp + 0, v.x);
    unsafeAtomicAdd(pp + 1, v.y);
    unsafeAtomicAdd(pp + 2, v.z);
    unsafeAtomicAdd(pp + 3, v.w);
}

__global__ void count_kernel(const int* __restrict__ batch, float* __restrict__ cnt, int N) {
    int i = blockIdx.x * blockDim.x + threadIdx.x;
    if (i < N) unsafeAtomicAdd(&cnt[batch[i]], 1.0f);
}

// ---------------- classifier: out[G,C] = (pooled/count) @ Wl + bl ----------------

__global__ void final_kernel(const float* __restrict__ pooled, const float* __restrict__ cnt,
                             const float* __restrict__ Wl, const float* __restrict__ bl,
                             float* __restrict__ out) {
    int idx = blockIdx.x * blockDim.x + threadIdx.x;
    if (idx >= Gg * Cc) return;
    int g = idx / Cc, c = idx % Cc;
    float inv = 1.0f / fmaxf(cnt[g], 1.0f);
    float s = 0.f;
    #pragma unroll 8
    for (int k = 0; k < Hh; ++k) s += pooled[g * Hh + k] * Wl[k * Cc + c];
    out[idx] = s * inv + bl[c];
}

// ---------------- host launcher ----------------

extern "C" void kernel_launch(void* const* d_in, const int* in_sizes, int n_in,
                              void* d_out, int out_size, void* d_ws, size_t ws_size,
                              hipStream_t stream) {
    (void)in_sizes; (void)n_in; (void)out_size; (void)ws_size;

    const float* x   = (const float*)d_in[0];
    const int*   ei  = (const int*)d_in[1];
    const int*   bat = (const int*)d_in[2];
    const float* W1  = (const float*)d_in[3];
    const float* b1  = (const float*)d_in[4];
    const float* W2  = (const float*)d_in[5];
    const float* b2  = (const float*)d_in[6];
    const float* W3  = (const float*)d_in[7];
    const float* b3  = (const float*)d_in[8];
    const float* Wl  = (const float*)d_in[9];
    const float* bl  = (const float*)d_in[10];
    float* out = (float*)d_out;

    const int* src = ei;        // edge_index[0]
    const int* dst = ei + Ee;   // edge_index[1]

    char* ws = (char*)d_ws;
    float* Y      = (float*)(ws + 0);            // 25,600,000 B
    float* AGG    = (float*)(ws + 25600000);     // 25,600,000 B
    int*   degc   = (int*)  (ws + 51200000);     //    200,000 B
    float* dinv   = (float*)(ws + 51400000);     //    200,000 B
    float* norm   = (float*)(ws + 51600000);     //  2,400,000 B
    float* pooled = (float*)(ws + 54000000);     //     32,768 B
    float* cnt    = (float*)(ws + 54032768);     //        256 B

    const size_t AGGB = (size_t)Nn * Hh * sizeof(float);

    // normalization coefficients
    hipMemsetAsync(degc, 0, (size_t)Nn * sizeof(int), stream);
    deg_count_kernel<<<(Ee + 255) / 256, 256, 0, stream>>>(dst, degc, Ee);
    dinv_kernel<<<(Nn + 255) / 256, 256, 0, stream>>>(degc, dinv, Nn);
    norm_kernel<<<(Ee + 255) / 256, 256, 0, stream>>>(src, dst, dinv, norm, Ee);

    const int gemmGrid  = Nn / (5 * 16);            // 625
    const int edgeGrid  = (Ee * 32) / 256;          // 75000
    const int fuseGrid  = (Nn * Hh) / 256;          // 25000

    // layer 1: h1 = relu(scatter(norm * (x@W1)) + b1)
    gemm_wmma_f32<<<gemmGrid, 256, 0, stream>>>(x, W1, Y);
    hipMemsetAsync(AGG, 0, AGGB, stream);
    edge_scatter_kernel<<<edgeGrid, 256, 0, stream>>>(Y, src, dst, norm, AGG, Ee);
    fuse_kernel<<<fuseGrid, 256, 0, stream>>>(AGG, Y, dinv, b1, 1);

    // layer 2
    gemm_wmma_f32<<<gemmGrid, 256, 0, stream>>>(AGG, W2, Y);
    hipMemsetAsync(AGG, 0, AGGB, stream);
    edge_scatter_kernel<<<edgeGrid, 256, 0, stream>>>(Y, src, dst, norm, AGG, Ee);
    fuse_kernel<<<fuseGrid, 256, 0, stream>>>(AGG, Y, dinv, b2, 1);

    // layer 3 (no relu)
    gemm_wmma_f32<<<gemmGrid, 256, 0, stream>>>(AGG, W3, Y);
    hipMemsetAsync(AGG, 0, AGGB, stream);
    edge_scatter_kernel<<<edgeGrid, 256, 0, stream>>>(Y, src, dst, norm, AGG, Ee);
    fuse_kernel<<<fuseGrid, 256, 0, stream>>>(AGG, Y, dinv, b3, 0);

    // mean pool + classifier
    hipMemsetAsync(pooled, 0, (size_t)Gg * Hh * sizeof(float) + (size_t)Gg * sizeof(float), stream);
    pool_kernel<<<(Nn * 32) / 256, 256, 0, stream>>>(AGG, bat, pooled);
    count_kernel<<<(Nn + 255) / 256, 256, 0, stream>>>(bat, cnt, Nn);
    final_kernel<<<(Gg * Cc + 255) / 256, 256, 0, stream>>>(pooled, cnt, Wl, bl, out);
}